// LinearAttention_21285857919698
// MI455X (gfx1250) — compile-verified
//
#include <hip/hip_runtime.h>

// ---------------- problem constants ----------------
#define CH      128
#define HEADS   4
#define DH      32
#define NMEM    4
#define GROUPS  32
#define HID     128
#define NB      2
#define NSPAT   65536          // D*H*W = 16*64*64
#define EPS     1e-5f
#define SCALE   0.17677669529663687f   // DH^-0.5

#define NC      32             // n-columns per tile chunk
#define NCHUNK  (NSPAT / NC)   // 2048 chunks per batch
#define LDSTR   36             // padded LDS row stride (16B-aligned rows, bank skew)

// ---------------- workspace layout (floats) ----------------
#define OFF_STATS1 0                       // NB*GROUPS*2        = 128
#define OFF_KMAX   128                     // NB*HEADS*DH        = 256
#define OFF_KZ     384                     // NB*HEADS*DH        = 256
#define OFF_CTX    640                     // NB*HEADS*DH*DH     = 8192
#define OFF_WEFF   8832                    // NB*CH*HID          = 32768
#define OFF_STATS2 41600                   // NB*GROUPS*2        = 128
#define OFF_WS     41728                   // NB*3*HID*CH        = 98304 (GN1-scaled qkv weights)
#define OFF_BETA   140032                  // NB*3*HID           = 768   (GN1 shift folded into rows)
#define OFF_Y      140800                  // NB*CH*NSPAT        = 16777216

typedef float v2f __attribute__((ext_vector_type(2)));
typedef float v8f __attribute__((ext_vector_type(8)));

// ---------------- helpers ----------------
__device__ inline void atomicMaxF(float* addr, float val) {
    unsigned int* ua = (unsigned int*)addr;
    unsigned int cur = *((volatile unsigned int*)ua);
    while (__uint_as_float(cur) < val) {
        unsigned int prev = atomicCAS(ua, cur, __float_as_uint(val));
        if (prev == cur) break;
        cur = prev;
    }
}

// Issue async LDS-DMA copies of one (b, n0) raw-x chunk (128 x NC floats, stride LDSTR).
// GLOBAL_LOAD_ASYNC_TO_LDS_B128, tracked by ASYNCcnt; no wait here (enables overlap).
__device__ inline void async_issue_x(const float* __restrict__ x, int b, int n0,
                                     float* lds) {
    for (int i = threadIdx.x; i < CH * (NC / 4); i += blockDim.x) {
        int c  = i >> 3;                 // NC/4 == 8 float4 per row
        int j4 = (i & 7) * 4;
        const float* src = x + ((size_t)b * CH + c) * NSPAT + n0 + j4;
        unsigned dst = (unsigned)(size_t)(lds + c * LDSTR + j4);  // low 32b = LDS offset
        asm volatile("global_load_async_to_lds_b128 %0, %1, off"
                     :: "v"(dst), "v"(src) : "memory");
    }
}
__device__ inline void async_wait() {
    asm volatile("s_wait_asynccnt 0x0" ::: "memory");
}

// Preload one 16-row A-slab (16 x 128, ld = CH) into per-lane registers:
// per k-step the lane's v2f = W[l15][kk+kofs .. +1]  (ISA 16x4 f32 A layout).
__device__ inline void load_areg(const float* __restrict__ Wg, v2f areg[32]) {
    const int lane = threadIdx.x & 31;
    const int l15  = lane & 15;
    const int kofs = (lane >> 4) << 1;
    #pragma unroll
    for (int kk4 = 0; kk4 < 32; ++kk4) {
        areg[kk4].x = Wg[l15 * CH + kk4 * 4 + kofs];
        areg[kk4].y = Wg[l15 * CH + kk4 * 4 + kofs + 1];
    }
}

// Wave-level GEMM with register-resident A: 16 rows x (NT*16) cols, K = 128.
template<int NT>
__device__ inline void gemm16r(const v2f areg[32], const float* __restrict__ lds,
                               v8f acc[NT]) {
    const int lane = threadIdx.x & 31;
    const int l15  = lane & 15;
    const int kofs = (lane >> 4) << 1;
    #pragma unroll
    for (int kk4 = 0; kk4 < 32; ++kk4) {
        int kk = kk4 * 4;
        #pragma unroll
        for (int t = 0; t < NT; ++t) {
            v2f bf;
            bf.x = lds[(kk + kofs)     * LDSTR + t * 16 + l15];
            bf.y = lds[(kk + kofs + 1) * LDSTR + t * 16 + l15];
            acc[t] = __builtin_amdgcn_wmma_f32_16x16x4_f32(
                false, areg[kk4], false, bf, (short)0, acc[t], false, false);
        }
    }
}

// 16x16 (K=NC) tile: ctx += Ek_slab @ V_slab^T, both operands in LDS row-major [row][k].
__device__ inline v8f gemm_ctx(const float* __restrict__ ekrow,
                               const float* __restrict__ vrow, v8f acc) {
    const int lane = threadIdx.x & 31;
    const int l15  = lane & 15;
    const int kofs = (lane >> 4) << 1;
    #pragma unroll
    for (int kk = 0; kk < NC; kk += 4) {
        v2f a, bf;
        a.x  = ekrow[l15 * LDSTR + kk + kofs];
        a.y  = ekrow[l15 * LDSTR + kk + kofs + 1];
        bf.x = vrow [l15 * LDSTR + kk + kofs];
        bf.y = vrow [l15 * LDSTR + kk + kofs + 1];
        acc = __builtin_amdgcn_wmma_f32_16x16x4_f32(
            false, a, false, bf, (short)0, acc, false, false);
    }
    return acc;
}

// ---------------- K1: GroupNorm1 stats ----------------
__global__ void k_stats1(const float* __restrict__ x, float* __restrict__ ws) {
    int blk = blockIdx.x;                // b*32 + g
    int b = blk >> 5, g = blk & 31;
    const float4* p4 = (const float4*)(x + ((size_t)b * CH + g * 4) * NSPAT);
    float s = 0.f, sq = 0.f;
    for (int i = threadIdx.x; i < (4 * NSPAT) / 4; i += blockDim.x) {
        float4 v = p4[i];
        s  += v.x + v.y + v.z + v.w;
        sq += v.x * v.x + v.y * v.y + v.z * v.z + v.w * v.w;
    }
    __shared__ float rs[256], rq[256];
    rs[threadIdx.x] = s; rq[threadIdx.x] = sq;
    __syncthreads();
    for (int off = 128; off > 0; off >>= 1) {
        if ((int)threadIdx.x < off) {
            rs[threadIdx.x] += rs[threadIdx.x + off];
            rq[threadIdx.x] += rq[threadIdx.x + off];
        }
        __syncthreads();
    }
    if (threadIdx.x == 0) {
        float cnt = 4.f * NSPAT;
        float mu  = rs[0] / cnt;
        float var = rq[0] / cnt - mu * mu;
        ws[OFF_STATS1 + blk * 2 + 0] = mu;
        ws[OFF_STATS1 + blk * 2 + 1] = rsqrtf(var + EPS);
    }
}

// ---------------- K2: fold GN1 affine into the qkv weights ----------------
__global__ void k_prep(const float* __restrict__ wqkv, const float* __restrict__ w1,
                       const float* __restrict__ b1, float* __restrict__ ws) {
    int t  = blockIdx.x * blockDim.x + threadIdx.x;
    int nt = gridDim.x * blockDim.x;
    for (int idx = t; idx < NB * 3 * HID * CH; idx += nt) {
        int b   = idx / (3 * HID * CH);
        int rem = idx - b * (3 * HID * CH);
        int o   = rem >> 7;
        int c   = rem & 127;
        float rsd = ws[OFF_STATS1 + (b * GROUPS + (c >> 2)) * 2 + 1];
        ws[OFF_WS + idx] = wqkv[o * CH + c] * (rsd * w1[c]);
    }
    for (int idx = t; idx < NB * 3 * HID; idx += nt) {
        int b = idx / (3 * HID);
        int o = idx - b * (3 * HID);
        float s = 0.f;
        for (int c = 0; c < CH; ++c) {
            float mu  = ws[OFF_STATS1 + (b * GROUPS + (c >> 2)) * 2 + 0];
            float rsd = ws[OFF_STATS1 + (b * GROUPS + (c >> 2)) * 2 + 1];
            float scv = rsd * w1[c];
            s += wqkv[o * CH + c] * (b1[c] - mu * scv);
        }
        ws[OFF_BETA + idx] = s;
    }
}

// ---------------- K3: init (zero accumulators, seed kmax from mem_k) ----------------
__global__ void k_init(const float* __restrict__ memkv, float* __restrict__ ws) {
    int t = blockIdx.x * blockDim.x + threadIdx.x;
    int nt = gridDim.x * blockDim.x;
    for (int i = t; i < NB * HEADS * DH * DH; i += nt) ws[OFF_CTX + i] = 0.f;
    for (int i = t; i < NB * HEADS * DH;      i += nt) ws[OFF_KZ  + i] = 0.f;
    for (int i = t; i < NB * GROUPS * 2;      i += nt) ws[OFF_STATS2 + i] = 0.f;
    for (int i = t; i < NB * HEADS * DH;      i += nt) {
        int hd = i & 127;
        const float* mk = memkv + hd * NMEM;     // mem_kv[0][h][d][:]
        float m = mk[0];
        #pragma unroll
        for (int j = 1; j < NMEM; ++j) m = fmaxf(m, mk[j]);
        ws[OFF_KMAX + i] = m;
    }
}

// ---------------- K4: global k-row max (k = Ws_k @ x + beta_k) ----------------
__global__ void k_kmax(const float* __restrict__ x, float* __restrict__ ws) {
    __shared__ float ldsd[2][CH * LDSTR];   // double-buffered async staging
    const int wave = threadIdx.x >> 5;
    const int lane = threadIdx.x & 31;
    const int l15  = lane & 15, half = lane >> 4;
    const int nblkpb = gridDim.x >> 1;
    const int b    = (blockIdx.x >= nblkpb) ? 1 : 0;
    const int blk0 = blockIdx.x - b * nblkpb;
    const float* wsW = ws + OFF_WS + (size_t)b * 3 * HID * CH;

    v2f areg[32];
    load_areg(wsW + (size_t)(CH + wave * 16) * CH, areg);   // k rows 128..255

    float rloc[8];
    #pragma unroll
    for (int i = 0; i < 8; ++i) rloc[i] = -3.0e38f;

    int cur = 0;
    if (blk0 < NCHUNK) async_issue_x(x, b, blk0 * NC, ldsd[0]);
    for (int ci = blk0; ci < NCHUNK; ci += nblkpb) {
        async_wait();
        __syncthreads();
        int cn = ci + nblkpb;
        if (cn < NCHUNK) async_issue_x(x, b, cn * NC, ldsd[cur ^ 1]);  // overlap DMA
        v8f acc[2];
        #pragma unroll
        for (int t = 0; t < 2; ++t)
            #pragma unroll
            for (int i = 0; i < 8; ++i) acc[t][i] = 0.f;
        gemm16r<2>(areg, ldsd[cur], acc);
        #pragma unroll
        for (int i = 0; i < 8; ++i)
            rloc[i] = fmaxf(rloc[i], fmaxf(acc[0][i], acc[1][i]));
        cur ^= 1;
    }
    #pragma unroll
    for (int i = 0; i < 8; ++i) {
        float m = rloc[i];
        #pragma unroll
        for (int msk = 1; msk < 16; msk <<= 1) m = fmaxf(m, __shfl_xor(m, msk, 16));
        if (l15 == 0) {
            int r = wave * 16 + half * 8 + i;           // 0..127 (= h*32+d)
            m += ws[OFF_BETA + b * 3 * HID + CH + r];   // fold row bias into max
            atomicMaxF(ws + OFF_KMAX + b * (HEADS * DH) + r, m);
        }
    }
}

// ---------------- K5: memory-kv contribution to ctx and Z ----------------
__global__ void k_memctx(const float* __restrict__ memkv, float* __restrict__ ws) {
    int t = threadIdx.x;
    if (t < NB * HEADS * DH) {
        int b = t >> 7, hd = t & 127;
        int h = hd >> 5, d = hd & 31;
        float kmax = ws[OFF_KMAX + t];
        const float* mk = memkv + hd * NMEM;
        float e[NMEM]; float z = 0.f;
        #pragma unroll
        for (int m = 0; m < NMEM; ++m) { e[m] = __expf(mk[m] - kmax); z += e[m]; }
        ws[OFF_KZ + t] += z;
        for (int ee = 0; ee < DH; ++ee) {
            const float* mv = memkv + (HEADS * DH + h * DH + ee) * NMEM;
            float s = 0.f;
            #pragma unroll
            for (int m = 0; m < NMEM; ++m) s += e[m] * mv[m];
            ws[OFF_CTX + (((size_t)b * HEADS + h) * DH + d) * DH + ee] += s;
        }
    }
}

// ---------------- K6: ctx += exp(k-kmax) @ v^T, Z += rowsum  (512 thr / 16 waves) ----------------
__global__ void k_ctx(const float* __restrict__ x, float* __restrict__ ws) {
    __shared__ float ldsx [CH * LDSTR];   // raw x chunk, then reused for V
    __shared__ float ldsEk[CH * LDSTR];   // exp(k - kmax)
    __shared__ float lec[CH];             // beta_k - kmax (exponent shift)
    __shared__ float lbv[CH];             // beta_v
    const int wave = threadIdx.x >> 5;    // 0..15
    const int lane = threadIdx.x & 31;
    const int l15  = lane & 15, half = lane >> 4;
    const int nblkpb = gridDim.x >> 1;
    const int b    = (blockIdx.x >= nblkpb) ? 1 : 0;
    const int blk0 = blockIdx.x - b * nblkpb;
    const float* wsW = ws + OFF_WS + (size_t)b * 3 * HID * CH;

    if (threadIdx.x < CH) {
        lec[threadIdx.x] = ws[OFF_BETA + b * 3 * HID + CH + threadIdx.x]
                         - ws[OFF_KMAX + b * CH + threadIdx.x];
        lbv[threadIdx.x] = ws[OFF_BETA + b * 3 * HID + 2 * CH + threadIdx.x];
    }

    const bool isk  = wave < 8;
    const int rbase = (wave & 7) * 16;             // row base within k- or v-block
    const int wrow  = CH + (isk ? 0 : CH) + rbase; // Ws rows: k=128..255, v=256..383

    v2f areg[32];
    load_areg(wsW + (size_t)wrow * CH, areg);

    // this wave's ctx tile: (h, mt, nt) = all 16 combos across 16 waves
    const int jh = wave >> 2, jmt = (wave >> 1) & 1, jnt = wave & 1;

    float zloc[8];
    v8f cacc;
    #pragma unroll
    for (int i = 0; i < 8; ++i) { zloc[i] = 0.f; cacc[i] = 0.f; }

    for (int ci = blk0; ci < NCHUNK; ci += nblkpb) {
        __syncthreads();                           // protect ldsx/ldsEk reuse
        async_issue_x(x, b, ci * NC, ldsx);
        async_wait();
        __syncthreads();
        v8f acc[2];
        #pragma unroll
        for (int t = 0; t < 2; ++t)
            #pragma unroll
            for (int i = 0; i < 8; ++i) acc[t][i] = 0.f;
        gemm16r<2>(areg, ldsx, acc);
        __syncthreads();                           // done reading ldsx
        #pragma unroll
        for (int nt = 0; nt < 2; ++nt)
            #pragma unroll
            for (int i = 0; i < 8; ++i) {
                int r   = rbase + half * 8 + i;
                int col = nt * 16 + l15;
                float val = acc[nt][i];
                if (isk) {
                    float ek = __expf(val + lec[r]);
                    ldsEk[r * LDSTR + col] = ek;
                    zloc[i] += ek;
                } else {
                    ldsx[r * LDSTR + col] = val + lbv[r];   // V
                }
            }
        __syncthreads();
        cacc = gemm_ctx(ldsEk + (jh * 32 + jmt * 16) * LDSTR,
                        ldsx  + (jh * 32 + jnt * 16) * LDSTR, cacc);
    }
    // flush Z
    if (isk) {
        #pragma unroll
        for (int i = 0; i < 8; ++i) {
            float z = zloc[i];
            #pragma unroll
            for (int msk = 1; msk < 16; msk <<= 1) z += __shfl_xor(z, msk, 16);
            if (l15 == 0)
                atomicAdd(ws + OFF_KZ + b * CH + rbase + half * 8 + i, z);
        }
    }
    // flush ctx partials
    #pragma unroll
    for (int i = 0; i < 8; ++i) {
        int d = jmt * 16 + half * 8 + i;
        int e = jnt * 16 + l15;
        atomicAdd(ws + OFF_CTX + (((size_t)b * HEADS + jh) * DH + d) * DH + e, cacc[i]);
    }
}

// ---------------- K7: W_eff = scale * (W_out ∘ ctx) / Z ----------------
__global__ void k_weff(const float* __restrict__ wout, float* __restrict__ ws) {
    int t  = blockIdx.x * blockDim.x + threadIdx.x;
    int nt = gridDim.x * blockDim.x;
    for (int idx = t; idx < NB * CH * HID; idx += nt) {
        int b  = idx >> 14;
        int o  = (idx >> 7) & 127;
        int hd = idx & 127;
        int h  = hd >> 5, d = hd & 31;
        float inv = SCALE / ws[OFF_KZ + b * CH + hd];
        const float* ctxp = ws + OFF_CTX + (((size_t)b * HEADS + h) * DH + d) * DH;
        const float* wo   = wout + o * HID + h * DH;
        float s = 0.f;
        #pragma unroll
        for (int e = 0; e < DH; ++e) s += wo[e] * ctxp[e];
        ws[OFF_WEFF + ((size_t)b * CH + o) * HID + hd] = s * inv;
    }
}

// ---------------- K8: q -> softmax_d -> y = W_eff@q_sm + b, + GN2 partial sums ----------------
__global__ void k_main(const float* __restrict__ x, const float* __restrict__ bout,
                       float* __restrict__ ws) {
    __shared__ float ldsd[2][CH * LDSTR];   // double-buffered async staging
    __shared__ float ldsq[CH * LDSTR];
    __shared__ float lbq[CH];
    const int wave = threadIdx.x >> 5;
    const int lane = threadIdx.x & 31;
    const int l15  = lane & 15, half = lane >> 4;
    const int nblkpb = gridDim.x >> 1;
    const int b    = (blockIdx.x >= nblkpb) ? 1 : 0;
    const int blk0 = blockIdx.x - b * nblkpb;
    const float* wsW  = ws + OFF_WS   + (size_t)b * 3 * HID * CH;
    const float* weff = ws + OFF_WEFF + (size_t)b * CH * HID;

    if (threadIdx.x < CH) lbq[threadIdx.x] = ws[OFF_BETA + b * 3 * HID + threadIdx.x];

    v2f aq[32], aw[32];
    load_areg(wsW  + (size_t)(wave * 16) * CH, aq);   // q rows 0..127
    load_areg(weff + (size_t)(wave * 16) * HID, aw);  // W_eff rows

    float sacc[2] = {0.f, 0.f}, qacc[2] = {0.f, 0.f};

    int cur = 0;
    if (blk0 < NCHUNK) async_issue_x(x, b, blk0 * NC, ldsd[0]);
    for (int ci = blk0; ci < NCHUNK; ci += nblkpb) {
        int n0 = ci * NC;
        async_wait();
        __syncthreads();
        int cn = ci + nblkpb;
        if (cn < NCHUNK) async_issue_x(x, b, cn * NC, ldsd[cur ^ 1]);   // overlap DMA
        v8f acc[2];
        #pragma unroll
        for (int t = 0; t < 2; ++t)
            #pragma unroll
            for (int i = 0; i < 8; ++i) acc[t][i] = 0.f;
        gemm16r<2>(aq, ldsd[cur], acc);
        #pragma unroll
        for (int nt = 0; nt < 2; ++nt)
            #pragma unroll
            for (int i = 0; i < 8; ++i) {
                int r = wave * 16 + half * 8 + i;
                ldsq[r * LDSTR + nt * 16 + l15] = acc[nt][i] + lbq[r];
            }
        __syncthreads();
        // softmax over d (32 rows per head) per column
        if (threadIdx.x < NC * HEADS) {
            int col = threadIdx.x & (NC - 1);
            int h   = threadIdx.x >> 5;
            float* qc = ldsq + h * DH * LDSTR + col;
            float mx = qc[0];
            #pragma unroll
            for (int d = 1; d < DH; ++d) mx = fmaxf(mx, qc[d * LDSTR]);
            float ssum = 0.f;
            #pragma unroll
            for (int d = 0; d < DH; ++d) {
                float e = __expf(qc[d * LDSTR] - mx);
                qc[d * LDSTR] = e; ssum += e;
            }
            float inv = 1.f / ssum;
            #pragma unroll
            for (int d = 0; d < DH; ++d) qc[d * LDSTR] *= inv;
        }
        __syncthreads();
        v8f acc2[2];
        #pragma unroll
        for (int t = 0; t < 2; ++t)
            #pragma unroll
            for (int i = 0; i < 8; ++i) acc2[t][i] = 0.f;
        gemm16r<2>(aw, ldsq, acc2);
        #pragma unroll
        for (int nt = 0; nt < 2; ++nt)
            #pragma unroll
            for (int i = 0; i < 8; ++i) {
                int r = wave * 16 + half * 8 + i;
                float val = acc2[nt][i] + bout[r];
                ws[OFF_Y + ((size_t)(b * CH + r)) * NSPAT + n0 + nt * 16 + l15] = val;
                int gi = i >> 2;                 // rows i<4 -> group g0, i>=4 -> g0+1
                sacc[gi] += val; qacc[gi] += val * val;
            }
        cur ^= 1;
    }
    #pragma unroll
    for (int gi = 0; gi < 2; ++gi) {
        float s = sacc[gi], q = qacc[gi];
        #pragma unroll
        for (int msk = 1; msk < 16; msk <<= 1) {
            s += __shfl_xor(s, msk, 16);
            q += __shfl_xor(q, msk, 16);
        }
        if (l15 == 0) {
            int g = wave * 4 + half * 2 + gi;
            atomicAdd(ws + OFF_STATS2 + (b * GROUPS + g) * 2 + 0, s);
            atomicAdd(ws + OFF_STATS2 + (b * GROUPS + g) * 2 + 1, q);
        }
    }
}

// ---------------- K9: GroupNorm2 apply ----------------
__global__ void k_final(const float* __restrict__ n2w, const float* __restrict__ n2b,
                        const float* __restrict__ ws, float* __restrict__ out) {
    int blk = blockIdx.x;                 // b*128 + c
    int b = blk >> 7, c = blk & 127;
    int g = c >> 2;
    float s   = ws[OFF_STATS2 + (b * GROUPS + g) * 2 + 0];
    float q   = ws[OFF_STATS2 + (b * GROUPS + g) * 2 + 1];
    float cnt = 4.f * NSPAT;
    float mu  = s / cnt;
    float rstd = rsqrtf(q / cnt - mu * mu + EPS);
    float sc = rstd * n2w[c];
    float sh = n2b[c] - mu * sc;
    const float4* yp = (const float4*)(ws + OFF_Y + (size_t)blk * NSPAT);
    float4* op = (float4*)(out + (size_t)blk * NSPAT);
    for (int i = threadIdx.x; i < NSPAT / 4; i += blockDim.x) {
        float4 v = yp[i];
        op[i] = make_float4(v.x * sc + sh, v.y * sc + sh, v.z * sc + sh, v.w * sc + sh);
    }
}

// ---------------- launch ----------------
extern "C" void kernel_launch(void* const* d_in, const int* in_sizes, int n_in,
                              void* d_out, int out_size, void* d_ws, size_t ws_size,
                              hipStream_t stream) {
    const float* x    = (const float*)d_in[0];
    const float* n1w  = (const float*)d_in[1];
    const float* n1b  = (const float*)d_in[2];
    const float* wqkv = (const float*)d_in[3];
    const float* mkv  = (const float*)d_in[4];
    const float* wout = (const float*)d_in[5];
    const float* bout = (const float*)d_in[6];
    const float* n2w  = (const float*)d_in[7];
    const float* n2b  = (const float*)d_in[8];
    float* out = (float*)d_out;
    float* ws  = (float*)d_ws;

    k_stats1<<<NB * GROUPS, 256, 0, stream>>>(x, ws);
    k_prep  <<<192, 256, 0, stream>>>(wqkv, n1w, n1b, ws);
    k_init  <<<32, 256, 0, stream>>>(mkv, ws);
    k_kmax  <<<512, 256, 0, stream>>>(x, ws);
    k_memctx<<<1, 256, 0, stream>>>(mkv, ws);
    k_ctx   <<<512, 512, 0, stream>>>(x, ws);
    k_weff  <<<64, 256, 0, stream>>>(wout, ws);
    k_main  <<<512, 256, 0, stream>>>(x, bout, ws);
    k_final <<<NB * CH, 256, 0, stream>>>(n2w, n2b, ws, out);
}